// CrossEmbed2GraphByProduct_28613072126233
// MI455X (gfx1250) — compile-verified
//
#include <hip/hip_runtime.h>

typedef __attribute__((ext_vector_type(2))) float v2f;
typedef __attribute__((ext_vector_type(4))) float v4f;
typedef __attribute__((ext_vector_type(8))) float v8f;

#define NB   512   // batches
#define RDIM 264   // ROIs (multiple of 8 -> row groups never straddle the boundary)
#define KDIM 32    // embedding dim
#define NS   9     // subnetworks
#define RP   272   // ROIs padded to 17 tiles of 16
#define LSTR 36    // LDS row stride (floats): 16B-aligned, conflict-free ds_load_b64
#define NT   17    // 16x16 tiles per dimension

__global__ __launch_bounds__(256) void gram_wmma_kernel(
    const float* __restrict__ emb,      // [NB, RDIM, KDIM]
    const int*   __restrict__ ends,     // [NS] cumulative block ends
    float* __restrict__ out_intra,      // [NB, RDIM, RDIM]
    float* __restrict__ out_inter,      // [NB, NS, NS]
    float* __restrict__ out_adj)        // [NB, RDIM, RDIM]
{
  __shared__ float lA[RP * LSTR];      // 39168 B: padded A tile (rows >= 264 are zero)
  __shared__ int   lseg[RDIM];         // segment id per ROI
  __shared__ float linter[NS * NS];    // inter-block partial sums
  __shared__ int   lends[NS];

  const int tid  = threadIdx.x;
  const int lane = tid & 31;
  const int wave = tid >> 5;           // 8 waves (wave32)
  const int b    = blockIdx.x;

  if (tid < NS)      lends[tid]  = ends[tid];
  if (tid < NS * NS) linter[tid] = 0.0f;

  // Stage A into LDS (float4 copies, zero-fill pad rows 264..271).
  const float* Ab = emb + (size_t)b * (RDIM * KDIM);
  for (int idx = tid; idx < RP * (KDIM / 4); idx += 256) {
    const int row = idx >> 3;
    const int c4  = (idx & 7) << 2;
    v4f v = {0.f, 0.f, 0.f, 0.f};
    if (row < RDIM) v = *(const v4f*)(Ab + row * KDIM + c4);
    *(v4f*)(&lA[row * LSTR + c4]) = v;
  }
  __syncthreads();

  if (tid < RDIM) {
    int s = 0;
#pragma unroll
    for (int j = 0; j < NS - 1; ++j)
      if (tid >= lends[j]) s = j + 1;
    lseg[tid] = s;
  }
  __syncthreads();

  const int m    = lane & 15;   // row/col within tile for this lane
  const int half = lane >> 4;   // K-pair selector per ISA 32-bit A layout

  // 17x17 = 289 tiles of G, round-robined over 8 waves.
  for (int t = wave; t < NT * NT; t += 8) {
    const int ti = t / NT;
    const int tj = t % NT;
    const float* arow = &lA[(ti * 16 + m) * LSTR];
    const float* brow = &lA[(tj * 16 + m) * LSTR];  // gram: B-frag pattern == A-frag pattern
    v8f acc = {0.f, 0.f, 0.f, 0.f, 0.f, 0.f, 0.f, 0.f};
#pragma unroll
    for (int kk = 0; kk < 8; ++kk) {
      const int kb = kk * 4 + half * 2;             // lanes 16-31 hold K=+2,+3
      v2f a  = *(const v2f*)(arow + kb);
      v2f bm = *(const v2f*)(brow + kb);
      acc = __builtin_amdgcn_wmma_f32_16x16x4_f32(
          /*neg_a=*/false, a, /*neg_b=*/false, bm,
          /*c_mod=*/(short)0, acc, /*reuse_a=*/false, /*reuse_b=*/false);
    }

    // Epilogue. C/D layout: VGPR v -> row r0+v with r0 = 16*ti + 8*half,
    // col c = 16*tj + (lane&15). RDIM % 8 == 0 => the 8-row group is either
    // fully in range or fully out: one guard, straight-line body.
    const int c  = tj * 16 + m;
    const int r0 = ti * 16 + 8 * half;
    if (c < RDIM && r0 < RDIM) {
      const int tcol = lseg[c];
      const int s0   = lseg[r0];
      const int s1   = lseg[r0 + 7];
      // All segment sizes >= 24 > 8: at most one boundary inside the group.
      // bnd = first row of segment s1 (or +inf if the group is single-segment).
      const int bnd = (s0 == s1) ? 0x7fffffff : lends[s1 - 1];

      float sum0 = 0.0f, sum1 = 0.0f;
      size_t off = (size_t)b * ((size_t)RDIM * RDIM) + (size_t)r0 * RDIM + c;
#pragma unroll
      for (int v = 0; v < 8; ++v) {
        const float val   = acc[v];
        const bool  lowseg = (r0 + v) < bnd;
        const int   sr     = lowseg ? s0 : s1;
        out_adj[off]   = val;
        out_intra[off] = (sr == tcol) ? val : 0.0f;
        if (lowseg) sum0 += val; else sum1 += val;
        off += RDIM;
      }
      atomicAdd(&linter[s0 * NS + tcol], sum0);          // ds_add_f32
      if (s1 != s0) atomicAdd(&linter[s1 * NS + tcol], sum1);
    }
  }
  __syncthreads();

  if (tid < NS * NS) {
    const int s  = tid / NS;
    const int t2 = tid % NS;
    const int ss = lends[s]  - (s  ? lends[s - 1]  : 0);
    const int st = lends[t2] - (t2 ? lends[t2 - 1] : 0);
    out_inter[(size_t)b * (NS * NS) + tid] = linter[tid] / (float)(ss * st);
  }
}

extern "C" void kernel_launch(void* const* d_in, const int* in_sizes, int n_in,
                              void* d_out, int out_size, void* d_ws, size_t ws_size,
                              hipStream_t stream) {
  const float* emb  = (const float*)d_in[0];
  const int*   ends = (const int*)d_in[1];

  float* out       = (float*)d_out;
  float* out_intra = out;                                       // [NB,R,R]
  float* out_inter = out + (size_t)NB * RDIM * RDIM;            // [NB,S,S]
  float* out_adj   = out_inter + (size_t)NB * NS * NS;          // [NB,R,R]

  gram_wmma_kernel<<<NB, 256, 0, stream>>>(emb, ends, out_intra, out_inter, out_adj);
}